// ReadoutLayer_51238959841811
// MI455X (gfx1250) — compile-verified
//
#include <hip/hip_runtime.h>
#include <math.h>

#define HEADS 4
#define HID 16
#define NEG_SLOPE 0.2f
#define IN_DIM 128

typedef __attribute__((ext_vector_type(2))) float v2f;
typedef __attribute__((ext_vector_type(8))) float v8f;

// ---------------------------------------------------------------- utilities
__device__ __forceinline__ void atomicMaxF(float* addr, float val) {
  // bit-trick float max; correct when initialized to -inf
  if (val >= 0.0f) atomicMax((int*)addr, __float_as_int(val));
  else             atomicMin((unsigned int*)addr, __float_as_uint(val));
}

__global__ void k_fill(float* __restrict__ p, float v, long long n) {
  long long i = (long long)blockIdx.x * blockDim.x + threadIdx.x;
  if (i < n) p[i] = v;
}

// ------------------------------------------------- WMMA fp32 GEMM core
// One wave32 computes one 16x16 tile of D = A[N,128] x B[128,KOUT].
// A-frag (16x4 f32): lanes 0-15 rows M, VGPR0=K0/K2 halves, VGPR1=K1/K3.
// B-frag (4x16 f32): lanes 0-15 cols N, VGPR0=K0/K2 halves, VGPR1=K1/K3.
// D (16x16 f32): VGPR v -> row v (lanes 0-15) / v+8 (lanes 16-31).
__device__ __forceinline__ v8f wmma_tile_16x16(const float* __restrict__ A,
                                               const float* __restrict__ B,
                                               int arow, int bcol, int kout, int half) {
  v8f acc = {0.f,0.f,0.f,0.f,0.f,0.f,0.f,0.f};
  const float* ap = A + (long long)arow * IN_DIM + half * 2;
  const float* bp = B + (long long)(half * 2) * kout + bcol;
  #pragma unroll 4
  for (int kk = 0; kk < IN_DIM; kk += 4) {
    v2f a, b;
    a.x = ap[0];
    a.y = ap[1];
    b.x = bp[0];
    b.y = bp[(long long)kout];
    acc = __builtin_amdgcn_wmma_f32_16x16x4_f32(false, a, false, b,
                                                (short)0, acc, false, false);
    ap += 4;
    bp += (long long)4 * kout;
  }
  return acc;
}

// h1 = x @ W1   (KOUT = 64)
__global__ void k_gemm_h1(const float* __restrict__ X, const float* __restrict__ W,
                          float* __restrict__ H1, int N) {
  const int KOUT = HEADS * HID;                    // 64
  const int ntn  = KOUT / 16;                      // 4 N-tiles
  int tile = blockIdx.x * (blockDim.x >> 5) + (threadIdx.x >> 5);
  int mt = tile / ntn, nt = tile % ntn;
  if (mt * 16 >= N) return;                        // wave-uniform guard
  int lane = threadIdx.x & 31;
  int half = lane >> 4, l16 = lane & 15;
  int arow = mt * 16 + l16; if (arow >= N) arow = N - 1;
  int bcol = nt * 16 + l16;
  v8f acc = wmma_tile_16x16(X, W, arow, bcol, KOUT, half);
  int rowBase = mt * 16 + half * 8;
  float* base = H1 + (long long)rowBase * KOUT + nt * 16 + l16;
  if (mt * 16 + 16 <= N) {                         // full tile: fast path
    #pragma unroll
    for (int v = 0; v < 8; ++v) base[v * KOUT] = acc[v];
  } else {                                         // ragged tail
    #pragma unroll
    for (int v = 0; v < 8; ++v)
      if (rowBase + v < N) base[v * KOUT] = acc[v];
  }
}

// x_ = att * relu(x @ Wemb + b_emb)   (KOUT = 128, fused gate epilogue)
__global__ void k_gemm_emb(const float* __restrict__ X, const float* __restrict__ W,
                           const float* __restrict__ bias, const float* __restrict__ att,
                           float* __restrict__ OUT, int N) {
  const int KOUT = IN_DIM;                         // 128
  const int ntn  = KOUT / 16;                      // 8 N-tiles
  int tile = blockIdx.x * (blockDim.x >> 5) + (threadIdx.x >> 5);
  int mt = tile / ntn, nt = tile % ntn;
  if (mt * 16 >= N) return;                        // wave-uniform guard
  int lane = threadIdx.x & 31;
  int half = lane >> 4, l16 = lane & 15;
  int arow = mt * 16 + l16; if (arow >= N) arow = N - 1;
  int bcol = nt * 16 + l16;
  v8f acc = wmma_tile_16x16(X, W, arow, bcol, KOUT, half);
  float bv = bias[nt * 16 + l16];
  int rowBase = mt * 16 + half * 8;
  float* base = OUT + (long long)rowBase * KOUT + nt * 16 + l16;
  const float* ab = att + rowBase;
  if (mt * 16 + 16 <= N) {                         // full tile: fast path
    #pragma unroll
    for (int v = 0; v < 8; ++v) {
      float r = acc[v] + bv;
      r = r > 0.0f ? r : 0.0f;
      base[v * KOUT] = ab[v] * r;
    }
  } else {                                         // ragged tail
    #pragma unroll
    for (int v = 0; v < 8; ++v) {
      if (rowBase + v < N) {
        float r = acc[v] + bv;
        r = r > 0.0f ? r : 0.0f;
        base[v * KOUT] = ab[v] * r;
      }
    }
  }
}

// ------------------------------------------------- GAT layer-1 helper kernels
__global__ void k_alphas(const float* __restrict__ h1, const float* __restrict__ aS,
                         const float* __restrict__ aD, float* __restrict__ alS,
                         float* __restrict__ alD, int N) {
  int i = blockIdx.x * blockDim.x + threadIdx.x;   // over N*HEADS
  if (i >= N * HEADS) return;
  int n = i >> 2, h = i & 3;
  const float* row = h1 + (long long)n * (HEADS * HID) + h * HID;
  float s = 0.f, d = 0.f;
  #pragma unroll
  for (int c = 0; c < HID; ++c) {
    s += row[c] * aS[h * HID + c];
    d += row[c] * aD[h * HID + c];
  }
  alS[i] = s; alD[i] = d;
}

__device__ __forceinline__ float leaky(float x) {
  return x >= 0.0f ? x : NEG_SLOPE * x;
}

__global__ void k_edge_max1(const int* __restrict__ src, const int* __restrict__ dst,
                            const float* __restrict__ alS, const float* __restrict__ alD,
                            float* __restrict__ m, long long EH) {
  long long i = (long long)blockIdx.x * blockDim.x + threadIdx.x;
  if (i >= EH) return;
  long long e = i >> 2; int h = (int)(i & 3);
  int s = src[e], d = dst[e];
  float ev = leaky(alS[(long long)s * 4 + h] + alD[(long long)d * 4 + h]);
  atomicMaxF(&m[(long long)d * 4 + h], ev);
}

__global__ void k_edge_sum1(const int* __restrict__ src, const int* __restrict__ dst,
                            const float* __restrict__ alS, const float* __restrict__ alD,
                            const float* __restrict__ m, float* __restrict__ den,
                            long long EH) {
  long long i = (long long)blockIdx.x * blockDim.x + threadIdx.x;
  if (i >= EH) return;
  long long e = i >> 2; int h = (int)(i & 3);
  int s = src[e], d = dst[e];
  float ev = leaky(alS[(long long)s * 4 + h] + alD[(long long)d * 4 + h]);
  float ex = __expf(ev - m[(long long)d * 4 + h]);
  atomicAdd(&den[(long long)d * 4 + h], ex);
}

__global__ void k_edge_agg1(const int* __restrict__ src, const int* __restrict__ dst,
                            const float* __restrict__ alS, const float* __restrict__ alD,
                            const float* __restrict__ m, const float* __restrict__ den,
                            const float* __restrict__ h1, float* __restrict__ agg,
                            long long EH) {
  long long i = (long long)blockIdx.x * blockDim.x + threadIdx.x;
  if (i >= EH) return;
  long long e = i >> 2; int h = (int)(i & 3);
  int s = src[e], d = dst[e];
  float ev = leaky(alS[(long long)s * 4 + h] + alD[(long long)d * 4 + h]);
  float ex = __expf(ev - m[(long long)d * 4 + h]);
  float alpha = ex / (den[(long long)d * 4 + h] + 1e-16f);
  const float* hs = h1 + (long long)s * (HEADS * HID) + h * HID;
  float* ad = agg + (long long)d * (HEADS * HID) + h * HID;
  #pragma unroll
  for (int c = 0; c < HID; ++c) atomicAdd(&ad[c], alpha * hs[c]);
}

__global__ void k_relu_bias(float* __restrict__ z, const float* __restrict__ b,
                            long long n) {
  long long i = (long long)blockIdx.x * blockDim.x + threadIdx.x;
  if (i >= n) return;
  float v = z[i] + b[i & 63];
  z[i] = v > 0.0f ? v : 0.0f;
}

// ------------------------------------------------- GAT layer-2 (H=1, C=1)
__global__ void k_h2(const float* __restrict__ z, const float* __restrict__ W2,
                     float* __restrict__ h2, int N) {
  int n = blockIdx.x * blockDim.x + threadIdx.x;
  if (n >= N) return;
  const float* row = z + (long long)n * 64;
  float acc = 0.f;
  #pragma unroll
  for (int k = 0; k < 64; ++k) acc += row[k] * W2[k];
  h2[n] = acc;
}

__global__ void k_edge_max2(const int* __restrict__ src, const int* __restrict__ dst,
                            const float* __restrict__ h2, const float* __restrict__ aS2,
                            const float* __restrict__ aD2, float* __restrict__ m,
                            long long E) {
  long long e = (long long)blockIdx.x * blockDim.x + threadIdx.x;
  if (e >= E) return;
  int s = src[e], d = dst[e];
  float ev = leaky(h2[s] * aS2[0] + h2[d] * aD2[0]);
  atomicMaxF(&m[d], ev);
}

__global__ void k_edge_sum2(const int* __restrict__ src, const int* __restrict__ dst,
                            const float* __restrict__ h2, const float* __restrict__ aS2,
                            const float* __restrict__ aD2, const float* __restrict__ m,
                            float* __restrict__ den, long long E) {
  long long e = (long long)blockIdx.x * blockDim.x + threadIdx.x;
  if (e >= E) return;
  int s = src[e], d = dst[e];
  float ev = leaky(h2[s] * aS2[0] + h2[d] * aD2[0]);
  atomicAdd(&den[d], __expf(ev - m[d]));
}

__global__ void k_edge_agg2(const int* __restrict__ src, const int* __restrict__ dst,
                            const float* __restrict__ h2, const float* __restrict__ aS2,
                            const float* __restrict__ aD2, const float* __restrict__ m,
                            const float* __restrict__ den, float* __restrict__ agg,
                            long long E) {
  long long e = (long long)blockIdx.x * blockDim.x + threadIdx.x;
  if (e >= E) return;
  int s = src[e], d = dst[e];
  float ev = leaky(h2[s] * aS2[0] + h2[d] * aD2[0]);
  float alpha = __expf(ev - m[d]) / (den[d] + 1e-16f);
  atomicAdd(&agg[d], alpha * h2[s]);
}

__global__ void k_att(float* __restrict__ att, const float* __restrict__ b2, int N) {
  int n = blockIdx.x * blockDim.x + threadIdx.x;
  if (n >= N) return;
  float v = att[n] + b2[0];
  att[n] = 1.0f / (1.0f + __expf(-v));
}

// ------------------------------------------------- pooling
__global__ void k_offsets(const int* __restrict__ len, int* __restrict__ off, int B) {
  if (threadIdx.x == 0 && blockIdx.x == 0) {
    int acc = 0;
    for (int b = 0; b < B; ++b) { off[b] = acc; acc += len[b]; }
  }
}

__global__ void k_pool(const float* __restrict__ xf, const int* __restrict__ len,
                       const int* __restrict__ off, float* __restrict__ out) {
  int b = blockIdx.x;
  int d = threadIdx.x;                 // 0..127
  int o = off[b], L = len[b];
  float mx = -1e30f, sum = 0.f;
  for (int i = 0; i < L; ++i) {
    float v = xf[(long long)(o + i) * IN_DIM + d];
    mx = fmaxf(mx, v);
    sum += v;
  }
  out[(long long)b * IN_DIM + d] = mx + sum / (float)L;
}

// ---------------------------------------------------------------- launcher
extern "C" void kernel_launch(void* const* d_in, const int* in_sizes, int n_in,
                              void* d_out, int out_size, void* d_ws, size_t ws_size,
                              hipStream_t stream) {
  const float* x     = (const float*)d_in[0];
  const int*   eidx  = (const int*)  d_in[2];
  const int*   lens  = (const int*)  d_in[3];
  const float* W1    = (const float*)d_in[4];
  const float* aS1   = (const float*)d_in[5];
  const float* aD1   = (const float*)d_in[6];
  const float* b1    = (const float*)d_in[7];
  const float* W2    = (const float*)d_in[8];
  const float* aS2   = (const float*)d_in[9];
  const float* aD2   = (const float*)d_in[10];
  const float* b2    = (const float*)d_in[11];
  const float* Wemb  = (const float*)d_in[12];
  const float* bemb  = (const float*)d_in[13];

  const int       N = in_sizes[0] / IN_DIM;       // 140000
  const long long E = (long long)in_sizes[2] / 2; // 1,820,000
  const int       B = in_sizes[3];                // 400
  const int* src = eidx;
  const int* dst = eidx + E;

  // workspace layout (floats)
  float* ws   = (float*)d_ws;
  float* h1   = ws;  ws += (long long)N * 64;
  float* alS1 = ws;  ws += (long long)N * 4;
  float* alD1 = ws;  ws += (long long)N * 4;
  float* m1   = ws;  ws += (long long)N * 4;
  float* den1 = ws;  ws += (long long)N * 4;
  float* agg1 = ws;  ws += (long long)N * 64;     // becomes z (in place)
  float* h2   = ws;  ws += N;
  float* m2   = ws;  ws += N;
  float* den2 = ws;  ws += N;
  float* agg2 = ws;  ws += N;                     // becomes att (in place)
  int*   offs = (int*)ws;

  float* out_pool = (float*)d_out;                 // [B,128]
  float* out_x    = (float*)d_out + (long long)B * IN_DIM; // [N,128]

  const int T = 256;
  auto blks = [](long long n, int t) { return (unsigned)((n + t - 1) / t); };

  // ---- init accumulators (every call: graph replays don't re-poison)
  k_fill<<<blks((long long)N*4, T), T, 0, stream>>>(m1,  -INFINITY, (long long)N*4);
  k_fill<<<blks((long long)N*4, T), T, 0, stream>>>(den1, 0.0f,     (long long)N*4);
  k_fill<<<blks((long long)N*64,T), T, 0, stream>>>(agg1, 0.0f,     (long long)N*64);
  k_fill<<<blks(N, T), T, 0, stream>>>(m2,  -INFINITY, N);
  k_fill<<<blks(N, T), T, 0, stream>>>(den2, 0.0f, N);
  k_fill<<<blks(N, T), T, 0, stream>>>(agg2, 0.0f, N);

  // ---- layer 1: projection (WMMA), attention coeffs, softmax-aggregate
  {
    long long tiles = ((long long)(N + 15) / 16) * (64 / 16);
    k_gemm_h1<<<blks(tiles, 8), 256, 0, stream>>>(x, W1, h1, N);
  }
  k_alphas<<<blks((long long)N*4, T), T, 0, stream>>>(h1, aS1, aD1, alS1, alD1, N);
  k_edge_max1<<<blks(E*4, T), T, 0, stream>>>(src, dst, alS1, alD1, m1, E*4);
  k_edge_sum1<<<blks(E*4, T), T, 0, stream>>>(src, dst, alS1, alD1, m1, den1, E*4);
  k_edge_agg1<<<blks(E*4, T), T, 0, stream>>>(src, dst, alS1, alD1, m1, den1, h1, agg1, E*4);
  k_relu_bias<<<blks((long long)N*64, T), T, 0, stream>>>(agg1, b1, (long long)N*64);

  // ---- layer 2 (scalar head) + sigmoid gate
  k_h2<<<blks(N, T), T, 0, stream>>>(agg1, W2, h2, N);
  k_edge_max2<<<blks(E, T), T, 0, stream>>>(src, dst, h2, aS2, aD2, m2, E);
  k_edge_sum2<<<blks(E, T), T, 0, stream>>>(src, dst, h2, aS2, aD2, m2, den2, E);
  k_edge_agg2<<<blks(E, T), T, 0, stream>>>(src, dst, h2, aS2, aD2, m2, den2, agg2, E);
  k_att<<<blks(N, T), T, 0, stream>>>(agg2, b2, N);

  // ---- embedding GEMM (WMMA) with fused relu+bias+gate -> x_ in d_out
  {
    long long tiles = ((long long)(N + 15) / 16) * (IN_DIM / 16);
    k_gemm_emb<<<blks(tiles, 8), 256, 0, stream>>>(x, Wemb, bemb, agg2, out_x, N);
  }

  // ---- ragged max+mean pooling
  k_offsets<<<1, 32, 0, stream>>>(lens, offs, B);
  k_pool<<<B, IN_DIM, 0, stream>>>(out_x, lens, offs, out_pool);
}